// GNN_Encoder_adj_mat_8486855376923
// MI455X (gfx1250) — compile-verified
//
#include <hip/hip_runtime.h>

typedef __attribute__((ext_vector_type(2))) float v2f;
typedef __attribute__((ext_vector_type(8))) float v8f;

#define EMB 64

// ---------------------------------------------------------------------------
// out[i] = cur[i] = concat(users, items)[i]      (float4 vectorized)
// ---------------------------------------------------------------------------
__global__ void init_concat_kernel(const float* __restrict__ users,
                                   const float* __restrict__ items,
                                   float* __restrict__ out,
                                   float* __restrict__ cur,
                                   int userElems, int totalElems) {
  int i4 = (blockIdx.x * blockDim.x + threadIdx.x) * 4;
  if (i4 >= totalElems) return;
  float4 v = (i4 < userElems) ? *(const float4*)(users + i4)
                              : *(const float4*)(items + (i4 - userElems));
  *(float4*)(out + i4) = v;
  *(float4*)(cur + i4) = v;
}

// ---------------------------------------------------------------------------
// out[i] = (out[i] + nxt[i]) * scale             (float4 vectorized)
// scale = 1 for intermediate layers, 0.25 fused into the last layer
// ---------------------------------------------------------------------------
__global__ void axpy_scale_kernel(float* __restrict__ out,
                                  const float* __restrict__ nxt,
                                  float scale, int totalElems) {
  int i4 = (blockIdx.x * blockDim.x + threadIdx.x) * 4;
  if (i4 >= totalElems) return;
  float4 a = *(const float4*)(out + i4);
  float4 b = *(const float4*)(nxt + i4);
  a.x = (a.x + b.x) * scale;
  a.y = (a.y + b.y) * scale;
  a.z = (a.z + b.z) * scale;
  a.w = (a.w + b.w) * scale;
  *(float4*)(out + i4) = a;
}

// ---------------------------------------------------------------------------
// COO SpMM:  y[row] += val * x[col]   for groups of 16 edges per wave.
//
// Formulated as D(16x64) = diag(vals_16) x Xg(16x64) using chained
// V_WMMA_F32_16X16X4_F32 (4 K-chunks of the diagonal per 16-column tile).
// Layouts per CDNA5 ISA 7.12.2:
//   A 16x4 f32 : lanes 0-15 M=0..15 {K0,K1}, lanes 16-31 M=0..15 {K2,K3}
//   B 4x16 f32 : VGPR0 = {row K0 | row K2}, VGPR1 = {row K1 | row K3}
//   D 16x16    : VGPR v = row v (lanes 0-15) / row v+8 (lanes 16-31), N = lane
// D rows are scattered with global_atomic_add_f32 (output stays f32, stays
// resident in the 192 MB L2).  Tail edges: indices clamped, diag value = 0,
// so EXEC remains all-ones as WMMA requires.
// ---------------------------------------------------------------------------
__global__ void spmm_wmma_kernel(const int*   __restrict__ rows,
                                 const int*   __restrict__ cols,
                                 const float* __restrict__ vals,
                                 const float* __restrict__ x,
                                 float*       __restrict__ y,
                                 int nEdges) {
  const int lane   = threadIdx.x & 31;
  const int m      = lane & 15;
  const bool hiHalf = lane >= 16;
  const int wavesPerBlock = blockDim.x >> 5;
  const int wave   = blockIdx.x * wavesPerBlock + (threadIdx.x >> 5);
  const int stride = gridDim.x * wavesPerBlock;
  const int nGroups = (nEdges + 15) >> 4;

  for (int g = wave; g < nGroups; g += stride) {
    const int base = g << 4;
    const int e    = base + m;
    const int eC   = (e < nEdges) ? e : (nEdges - 1);

    const float v   = (e < nEdges) ? vals[eC] : 0.0f;   // diag entry for row m
    const int myRow = rows[eC];
    const int myCol = cols[eC];

    // B source columns per K-chunk: lane-half<16 supplies K rows {4c,4c+1},
    // lane-half>=16 supplies {4c+2,4c+3}.
    int colK[8];
#pragma unroll
    for (int c = 0; c < 4; ++c) {
      const int kb0 = 4 * c + (hiHalf ? 2 : 0);
      colK[2 * c]     = __shfl(myCol, kb0, 32);
      colK[2 * c + 1] = __shfl(myCol, kb0 + 1, 32);
    }

    // Destination row for each D VGPR held by this lane.
    int rowD[8];
#pragma unroll
    for (int vv = 0; vv < 8; ++vv)
      rowD[vv] = __shfl(myRow, hiHalf ? (vv + 8) : vv, 32);

    // Diagonal A, sliced into 4 chunks of K=4.
    v2f A[4];
#pragma unroll
    for (int c = 0; c < 4; ++c) {
      const int ka0 = 4 * c + (hiHalf ? 2 : 0);
      A[c].x = (m == ka0)     ? v : 0.0f;
      A[c].y = (m == ka0 + 1) ? v : 0.0f;
    }

    // 4 N-tiles of 16 columns covering EMB=64.
#pragma unroll
    for (int t = 0; t < 4; ++t) {
      v8f acc = {};
#pragma unroll
      for (int c = 0; c < 4; ++c) {
        v2f B;
        B.x = x[colK[2 * c]     * EMB + t * 16 + m];
        B.y = x[colK[2 * c + 1] * EMB + t * 16 + m];
        acc = __builtin_amdgcn_wmma_f32_16x16x4_f32(
            /*neg_a=*/false, A[c], /*neg_b=*/false, B,
            /*c_mod=*/(short)0, acc, /*reuse_a=*/false, /*reuse_b=*/false);
      }
#pragma unroll
      for (int vv = 0; vv < 8; ++vv)
        atomicAdd(&y[rowD[vv] * EMB + t * 16 + m], acc[vv]);
    }
  }
}

// ---------------------------------------------------------------------------
extern "C" void kernel_launch(void* const* d_in, const int* in_sizes, int n_in,
                              void* d_out, int out_size, void* d_ws, size_t ws_size,
                              hipStream_t stream) {
  const float* users = (const float*)d_in[0];
  const float* items = (const float*)d_in[1];
  const int*   erow  = (const int*)d_in[2];
  const int*   ecol  = (const int*)d_in[3];
  const float* evals = (const float*)d_in[4];
  float* out = (float*)d_out;

  const int userElems  = in_sizes[0];
  const int itemElems  = in_sizes[1];
  const int nEdges     = in_sizes[2];
  const int totalElems = userElems + itemElems;   // N * 64

  const size_t bufBytes = (size_t)totalElems * sizeof(float);
  if (ws_size < 2 * bufBytes) return;
  float* cur = (float*)d_ws;
  float* nxt = (float*)((char*)d_ws + bufBytes);

  const int ethreads = (totalElems + 3) / 4;
  const dim3 eb(256), eg((ethreads + 255) / 256);
  init_concat_kernel<<<eg, eb, 0, stream>>>(users, items, out, cur,
                                            userElems, totalElems);

  const int nGroups    = (nEdges + 15) / 16;
  const int spmmBlocks = (nGroups + 7) / 8;       // 8 waves / 256-thread block

  for (int layer = 0; layer < 3; ++layer) {
    hipMemsetAsync(nxt, 0, bufBytes, stream);
    spmm_wmma_kernel<<<spmmBlocks, 256, 0, stream>>>(erow, ecol, evals,
                                                     cur, nxt, nEdges);
    const float scale = (layer == 2) ? 0.25f : 1.0f;
    axpy_scale_kernel<<<eg, eb, 0, stream>>>(out, nxt, scale, totalElems);
    float* tmp = cur; cur = nxt; nxt = tmp;       // ping-pong
  }
}